// SpatialAttention_36532991820024
// MI455X (gfx1250) — compile-verified
//
#include <hip/hip_runtime.h>

typedef __attribute__((ext_vector_type(2))) float v2f;
typedef __attribute__((ext_vector_type(8))) float v8f;

#define B_  8
#define N_  16384
#define K_  32
#define EPS_ 1e-5f

// ---- VALU-only cross-lane helpers (no ds_bpermute) ----

// DPP move with compile-time control code.
template<int CTRL>
__device__ __forceinline__ float dppmov(float v) {
    int r = __builtin_amdgcn_update_dpp(0, __builtin_bit_cast(int, v),
                                        CTRL, 0xF, 0xF, true);
    return __builtin_bit_cast(float, r);
}

// lane ^ 16 exchange: v_permlanex16_b32 with identity nibble pattern.
__device__ __forceinline__ float permx16(float v) {
    int i = __builtin_bit_cast(int, v);
    int r = __builtin_amdgcn_permlanex16(i, i, 0x76543210u, 0xFEDCBA98u,
                                         false, false);
    return __builtin_bit_cast(float, r);
}

#define DPP_QUAD_SWAP1   0xB1   // quad_perm(1,0,3,2)
#define DPP_QUAD_SWAP2   0x4E   // quad_perm(2,3,0,1)
#define DPP_ROW_HMIRROR  0x141  // row_half_mirror
#define DPP_ROW_MIRROR   0x140  // row_mirror

// Full wave32 reductions; result broadcast to all lanes.
__device__ __forceinline__ float wave_sum(float v) {
    v += dppmov<DPP_QUAD_SWAP1>(v);
    v += dppmov<DPP_QUAD_SWAP2>(v);
    v += dppmov<DPP_ROW_HMIRROR>(v);
    v += dppmov<DPP_ROW_MIRROR>(v);
    v += permx16(v);
    return v;
}
__device__ __forceinline__ float wave_max(float v) {
    v = fmaxf(v, dppmov<DPP_QUAD_SWAP1>(v));
    v = fmaxf(v, dppmov<DPP_QUAD_SWAP2>(v));
    v = fmaxf(v, dppmov<DPP_ROW_HMIRROR>(v));
    v = fmaxf(v, dppmov<DPP_ROW_MIRROR>(v));
    v = fmaxf(v, permx16(v));
    return v;
}

// One wave (32 lanes) per point (b,n); lane = neighbor k.
__global__ __launch_bounds__(256)
void spatial_attn_wmma_kernel(
    const float* __restrict__ xyz,        // [B,N,3]
    const float* __restrict__ intensity,  // [B,3,N]
    const int*   __restrict__ indices,    // [B,N,K]
    const float* __restrict__ w1,         // [3,3]
    const float* __restrict__ b1,         // [3]
    const float* __restrict__ gamma,      // [3]
    const float* __restrict__ beta,       // [3]
    const float* __restrict__ mean,       // [3]
    const float* __restrict__ var,        // [3]
    const float* __restrict__ w2,         // [1,3]
    const float* __restrict__ b2,         // [1]
    float* __restrict__ out)              // [B,3,N]
{
    const int tid  = blockIdx.x * blockDim.x + threadIdx.x;
    const int lane = threadIdx.x & 31;
    const int wave = tid >> 5;                 // point id in [0, B*N)
    const int b    = wave / N_;
    const int n    = wave - b * N_;

    // ---- fold BN into conv1 (uniform -> scalar loads / SALU) ----
    const float sc0 = gamma[0] * rsqrtf(var[0] + EPS_);
    const float sc1 = gamma[1] * rsqrtf(var[1] + EPS_);
    const float sc2 = gamma[2] * rsqrtf(var[2] + EPS_);
    const float W00 = w1[0]*sc0, W01 = w1[1]*sc0, W02 = w1[2]*sc0;
    const float W10 = w1[3]*sc1, W11 = w1[4]*sc1, W12 = w1[5]*sc1;
    const float W20 = w1[6]*sc2, W21 = w1[7]*sc2, W22 = w1[8]*sc2;
    const float be0 = (b1[0]-mean[0])*sc0 + beta[0];
    const float be1 = (b1[1]-mean[1])*sc1 + beta[1];
    const float be2 = (b1[2]-mean[2])*sc2 + beta[2];
    const float u0 = w2[0], u1 = w2[1], u2 = w2[2], ub = b2[0];

    // ---- gathers (the real bottleneck; indices load is coalesced) ----
    const long long ibase = (long long)wave * K_;
    const int idx = indices[ibase + lane];

    const float* xb = xyz + (long long)b * N_ * 3;
    const float cx = xb[n*3+0], cy = xb[n*3+1], cz = xb[n*3+2];
    const float px = xb[idx*3+0] - cx;
    const float py = xb[idx*3+1] - cy;
    const float pz = xb[idx*3+2] - cz;

    // gaussian features per lane
    const float g0 = __expf(-2.f * px * px);
    const float g1 = __expf(-2.f * py * py);
    const float g2 = __expf(-2.f * pz * pz);

    const float* ib = intensity + (long long)b * 3 * N_;
    const float i0 = ib[idx];
    const float i1 = ib[N_ + idx];
    const float i2 = ib[2*N_ + idx];

    // ---- 3x3 conv as V_WMMA_F32_16X16X4_F32 (D = A(16x4) x B(4x16)) ----
    // A layout: VGPR0 = {K=0 | K=2}, VGPR1 = {K=1 | K=3} over lane halves; rows M = lane&15.
    v2f a;
    a.x = (lane < 16)
            ? (lane == 0 ? W00 : lane == 1 ? W10 : lane == 2 ? W20 : 0.f)
            : (lane == 16 ? W02 : lane == 17 ? W12 : lane == 18 ? W22 : 0.f);
    a.y = (lane < 16)
            ? (lane == 0 ? W01 : lane == 1 ? W11 : lane == 2 ? W21 : 0.f)
            : 0.f;

    // B layout mirrors A: VGPR0 = {row0 | row2}, VGPR1 = {row1 | row3(=0)}.
    // All marshaling moves are lane^16 exchanges -> v_permlanex16_b32.
    const float g2x = permx16(g2);   // upper half reads lower-half g2
    const float g0x = permx16(g0);   // lower half reads upper-half g0
    const float g1x = permx16(g1);   // lower half reads upper-half g1

    v2f bm1, bm2;
    bm1.x = (lane < 16) ? g0  : g2x;  bm1.y = (lane < 16) ? g1  : 0.f;
    bm2.x = (lane < 16) ? g0x : g2;   bm2.y = (lane < 16) ? g1x : 0.f;

    v8f dz = {};
    v8f d1 = __builtin_amdgcn_wmma_f32_16x16x4_f32(false, a, false, bm1,
                                                   (short)0, dz, false, false);
    v8f d2 = __builtin_amdgcn_wmma_f32_16x16x4_f32(false, a, false, bm2,
                                                   (short)0, dz, false, false);

    // D rows 0..2 live in accumulator VGPRs 0..2 of lanes 0..15 (N = lane).
    // Elements 16..31 sit in lanes 0..15 of d2 -> route up via lane^16.
    const float r0 = permx16((float)d2[0]);
    const float r1 = permx16((float)d2[1]);
    const float r2 = permx16((float)d2[2]);
    float h0 = (lane < 16) ? (float)d1[0] : r0;
    float h1 = (lane < 16) ? (float)d1[1] : r1;
    float h2 = (lane < 16) ? (float)d1[2] : r2;

    // bias(+folded BN) + ReLU, then 3->1 conv
    h0 = fmaxf(h0 + be0, 0.f);
    h1 = fmaxf(h1 + be1, 0.f);
    h2 = fmaxf(h2 + be2, 0.f);
    const float logit = fmaf(u0, h0, fmaf(u1, h1, fmaf(u2, h2, ub)));

    // ---- softmax over K: VALU-only wave32 reductions ----
    const float m = wave_max(logit);
    const float e = __expf(logit - m);
    const float s = wave_sum(e);
    const float wgt = e * __frcp_rn(s);

    // ---- weighted reduce of gathered intensities ----
    const float o0 = wave_sum(wgt * i0);
    const float o1 = wave_sum(wgt * i1);
    const float o2 = wave_sum(wgt * i2);

    if (lane == 0) {
        float* ob = out + (long long)b * 3 * N_;
        ob[n]        = o0;
        ob[N_ + n]   = o1;
        ob[2*N_ + n] = o2;
    }
}

extern "C" void kernel_launch(void* const* d_in, const int* in_sizes, int n_in,
                              void* d_out, int out_size, void* d_ws, size_t ws_size,
                              hipStream_t stream) {
    (void)in_sizes; (void)n_in; (void)d_ws; (void)ws_size; (void)out_size;
    const float* xyz       = (const float*)d_in[0];
    const float* intensity = (const float*)d_in[1];
    const int*   indices   = (const int*)  d_in[2];
    const float* w1        = (const float*)d_in[3];
    const float* b1        = (const float*)d_in[4];
    const float* gamma     = (const float*)d_in[5];
    const float* beta      = (const float*)d_in[6];
    const float* mean      = (const float*)d_in[7];
    const float* var       = (const float*)d_in[8];
    const float* w2        = (const float*)d_in[9];
    const float* b2        = (const float*)d_in[10];
    float* out             = (float*)d_out;

    const int total_threads = B_ * N_ * 32;   // one wave per point
    const int block = 256;                    // 8 wave32s per block
    const int grid  = total_threads / block;  // 16384, exact

    spatial_attn_wmma_kernel<<<grid, block, 0, stream>>>(
        xyz, intensity, indices, w1, b1, gamma, beta, mean, var, w2, b2, out);
}